// SlowROIPool_20641612825007
// MI455X (gfx1250) — compile-verified
//
#include <hip/hip_runtime.h>
#include <stdint.h>

// Adaptive ROI max-pool (SlowROIPool) for MI455X / gfx1250.
// Data-movement bound (no matmul structure -> no WMMA). Uses the CDNA5
// Tensor Data Mover (tensor_load_to_lds + s_wait_tensorcnt) to DMA 2D ROI
// crops into LDS, double-buffered, and ds_max_u32 LDS atomics for the
// segmented width-pool.

#define OUTB 7
#define CT 32                 // channels per tkey tile
#define NG 2                  // channels staged per TDM group (double-buffered)
#define NWAVES 8
#define WG 256
#define HMAX 56
#define CROPMAX (56 * 56)     // max crop elements per channel
#define NEG_INF_KEY 0x007FFFFFu

typedef uint32_t u32x4 __attribute__((ext_vector_type(4)));
typedef int      i32x4 __attribute__((ext_vector_type(4)));
typedef int      i32x8 __attribute__((ext_vector_type(8)));

// Monotone bijection float -> uint32 so unsigned max == float max (no NaNs here).
__device__ __forceinline__ unsigned key_of(float x) {
    unsigned b = __float_as_uint(x);
    return (b & 0x80000000u) ? ~b : (b | 0x80000000u);
}
__device__ __forceinline__ float val_of(unsigned k) {
    unsigned b = (k & 0x80000000u) ? (k & 0x7FFFFFFFu) : ~k;
    return __uint_as_float(b);
}

__device__ __forceinline__ unsigned lds_byte_of(const void* p) {
    return (unsigned)(uintptr_t)p;   // flat LDS aperture: low 32 bits = LDS addr
}

// Issue one TDM 2D-tile load: Lh x Lw elements (f32), row stride W, into LDS.
// D# built per CDNA5 ISA ch.8: group0 (count=1, lds_addr, global_addr, type=2),
// group1 (data_size=4B, tensor dims, tile dims, dim0 stride). Groups 2/3 are
// zero-filled (tile_dim2==0 in group1 marks the tile as 2D, so they're unused).
// clang-23 (amdgpu-toolchain) 6-arg builtin form.
__device__ __forceinline__ void tdm_load_2d(unsigned lds_byte, uint64_t gaddr,
                                            int Lw, int Lh, int W) {
    u32x4 g0;
    g0[0] = 1u;                                                   // count=1 (valid)
    g0[1] = lds_byte;                                             // lds_addr
    g0[2] = (uint32_t)gaddr;                                      // global_addr[31:0]
    g0[3] = ((uint32_t)(gaddr >> 32) & 0x01FFFFFFu) | (2u << 30); // addr[56:32], type=2

    i32x8 g1;
    g1[0] = (int)0x00020000u;                                     // wg_mask=0, data_size=4B
    g1[1] = (int)(((uint32_t)Lw & 0xFFFFu) << 16);                // tensor_dim0[15:0]
    g1[2] = (int)((((uint32_t)Lw >> 16) & 0xFFFFu) |
                  (((uint32_t)Lh & 0xFFFFu) << 16));              // dim0 hi | tensor_dim1 lo
    g1[3] = (int)((((uint32_t)Lh >> 16) & 0xFFFFu) |
                  ((uint32_t)Lw << 16));                          // dim1 hi | tile_dim0=Lw
    g1[4] = (int)((uint32_t)Lh & 0xFFFFu);                        // tile_dim1=Lh, tile_dim2=0
    g1[5] = (int)(uint32_t)W;                                     // tensor_dim0_stride lo
    g1[6] = 0;                                                    // stride0 hi | stride1 lo
    g1[7] = 0;

    i32x4 g2 = {0, 0, 0, 0};
    i32x4 g3 = {0, 0, 0, 0};
    i32x8 g4 = {0, 0, 0, 0, 0, 0, 0, 0};

    __builtin_amdgcn_tensor_load_to_lds(g0, g1, g2, g3, g4, 0);
}

struct RoiInfo {
    int x1, Lw, y1, Lh, img;
    int sx[OUTB], ex[OUTB];     // absolute width-bin bounds [sx, ex)
    int rsy[OUTB], rey[OUTB];   // crop-relative height-bin bounds
};

__global__ __launch_bounds__(WG)
void roipool_tdm_kernel(const float* __restrict__ images,
                        const float* __restrict__ rois,
                        const int*   __restrict__ roi_idx,
                        float*       __restrict__ out,
                        int C, int H, int W) {
    extern __shared__ unsigned char dynsmem[];
    unsigned* tkey    = (unsigned*)dynsmem;                               // [CT][HMAX][OUTB]
    float*    cropbuf = (float*)(dynsmem + (size_t)CT * HMAX * OUTB * 4); // [2][NG][CROPMAX]

    __shared__ RoiInfo ri;

    const int n    = blockIdx.x;
    const int tid  = threadIdx.x;
    const int lane = tid & 31;
    const int wave = tid >> 5;

    if (tid == 0) {
        float rx1 = rois[n * 4 + 0], ry1 = rois[n * 4 + 1];
        float rx2 = rois[n * 4 + 2], ry2 = rois[n * 4 + 3];
        int x1 = (int)floorf(rx1 * (float)W);
        int x2 = (int)ceilf (rx2 * (float)W);
        int y1 = (int)floorf(ry1 * (float)H);
        int y2 = (int)ceilf (ry2 * (float)H);
        int Lw = x2 - x1, Lh = y2 - y1;
        ri.x1 = x1; ri.Lw = Lw; ri.y1 = y1; ri.Lh = Lh;
        ri.img = roi_idx[n];
        for (int j = 0; j < OUTB; ++j) {
            ri.sx[j]  = x1 + (j * Lw) / OUTB;
            ri.ex[j]  = x1 + ((j + 1) * Lw + OUTB - 1) / OUTB;  // ceil
            ri.rsy[j] = (j * Lh) / OUTB;
            ri.rey[j] = ((j + 1) * Lh + OUTB - 1) / OUTB;       // ceil
        }
    }
    __syncthreads();

    const int x1 = ri.x1, Lw = ri.Lw, y1 = ri.y1, Lh = ri.Lh, bimg = ri.img;

    // Per-lane width-bin membership masks (fixed for the whole ROI; a column
    // can belong to up to 2 adjacent bins under the adaptive-pool rule).
    unsigned maskA = 0, maskB = 0;
    {
        const int xa = x1 + lane, xb = xa + 32;
        #pragma unroll
        for (int j = 0; j < OUTB; ++j) {
            if (xa >= ri.sx[j] && xa < ri.ex[j]) maskA |= 1u << j;
            if (xb >= ri.sx[j] && xb < ri.ex[j]) maskB |= 1u << j;
        }
        if (lane >= Lw)      maskA = 0;
        if (lane + 32 >= Lw) maskB = 0;
    }

    const size_t plane = (size_t)H * W;
    const float* img0  = images + (size_t)bimg * C * plane + (size_t)y1 * W + x1;
    const int ngroups  = CT / NG;

    for (int ct = 0; ct < C; ct += CT) {
        // ---- init width-pool accumulators to key(-inf) ----
        for (int i = tid; i < CT * HMAX * OUTB; i += WG) tkey[i] = NEG_INF_KEY;

        // ---- prologue: TDM group 0 -> buffer 0 ----
        if (tid == 0) {
            for (int k = 0; k < NG; ++k)
                tdm_load_2d(lds_byte_of(&cropbuf[k * CROPMAX]),
                            (uint64_t)(uintptr_t)(img0 + (size_t)(ct + k) * plane),
                            Lw, Lh, W);
        }
        __syncthreads();

        for (int g = 0; g < ngroups; ++g) {
            const int buf = g & 1;
            if (tid == 0) {
                if (g + 1 < ngroups) {
                    // issue next group into the other buffer, then wait for
                    // the current group's NG in-order TDM ops to complete
                    for (int k = 0; k < NG; ++k)
                        tdm_load_2d(lds_byte_of(&cropbuf[((g + 1) & 1) * NG * CROPMAX + k * CROPMAX]),
                                    (uint64_t)(uintptr_t)(img0 + (size_t)(ct + (g + 1) * NG + k) * plane),
                                    Lw, Lh, W);
                    __builtin_amdgcn_s_wait_tensorcnt(NG);
                } else {
                    __builtin_amdgcn_s_wait_tensorcnt(0);
                }
            }
            __syncthreads();   // group g crop data now visible to all waves

            // ---- Pass A: width-pool the staged crops (ds_max_u32) ----
            for (int k = 0; k < NG; ++k) {
                const int cl = g * NG + k;
                const float* crop = &cropbuf[buf * NG * CROPMAX + k * CROPMAX];
                unsigned* trow0 = &tkey[cl * HMAX * OUTB];
                for (int rh = wave; rh < Lh; rh += NWAVES) {
                    const float* row = crop + rh * Lw;
                    unsigned* trow = trow0 + rh * OUTB;
                    if (maskA) {
                        const unsigned kk = key_of(row[lane]);
                        unsigned m = maskA;
                        while (m) { int j = __builtin_ctz(m); m &= m - 1; atomicMax(&trow[j], kk); }
                    }
                    if (maskB) {
                        const unsigned kk = key_of(row[lane + 32]);
                        unsigned m = maskB;
                        while (m) { int j = __builtin_ctz(m); m &= m - 1; atomicMax(&trow[j], kk); }
                    }
                }
            }
            __syncthreads();   // consumers done before this buffer is reused
        }

        // ---- Pass B: height-pool, write [N,C,7,7] ----
        for (int item = tid; item < CT * OUTB * OUTB; item += WG) {
            const int cl = item / (OUTB * OUTB);
            const int r  = item - cl * (OUTB * OUTB);
            const int iy = r / OUTB;
            const int jx = r - iy * OUTB;
            unsigned m = NEG_INF_KEY;
            const int re = ri.rey[iy];
            for (int rh = ri.rsy[iy]; rh < re; ++rh) {
                const unsigned v = tkey[(cl * HMAX + rh) * OUTB + jx];
                m = (v > m) ? v : m;
            }
            out[(((size_t)n * C + (ct + cl)) * OUTB + iy) * OUTB + jx] = val_of(m);
        }
        __syncthreads();
    }
}

extern "C" void kernel_launch(void* const* d_in, const int* in_sizes, int n_in,
                              void* d_out, int out_size, void* d_ws, size_t ws_size,
                              hipStream_t stream) {
    const float* images  = (const float*)d_in[0];
    const float* rois    = (const float*)d_in[1];
    const int*   roi_idx = (const int*)d_in[2];
    float*       out     = (float*)d_out;

    const int N = in_sizes[2];          // number of ROIs
    const int C = 128, H = 56, W = 56;  // fixed by the reference setup

    const size_t dyn_bytes = (size_t)CT * HMAX * OUTB * 4        // tkey
                           + (size_t)2 * NG * CROPMAX * 4;       // crop double-buffer

    roipool_tdm_kernel<<<dim3(N), dim3(WG), dyn_bytes, stream>>>(
        images, rois, roi_idx, out, C, H, W);
}